// GCN_1872605741509
// MI455X (gfx1250) — compile-verified
//
#include <hip/hip_runtime.h>

typedef float v2f __attribute__((ext_vector_type(2)));
typedef float v8f __attribute__((ext_vector_type(8)));

#define IN_CH   128
#define OUT_CH  16
#define WSTRIDE 130   // padded LDS stride for 16x128 weight tiles (bank-conflict free)
#define HSTRIDE 18    // padded LDS stride for 16x16 tiles (18j mod 64 distinct for j=0..15)
#define NEG_SLOPE 0.01f

__device__ __forceinline__ float leaky(float h) {
    return h >= 0.0f ? h : NEG_SLOPE * h;
}

// ---------------------------------------------------------------------------
// Kernel 1: xw[n, j] = sum_k x[n,k] * w[j,k]    (x: n_nodes x 128, w: 16 x 128)
// One wave per 16-node tile, V_WMMA_F32_16X16X4_F32, K looped 128/4 = 32 steps.
// x is a 256MB single-pass stream -> non-temporal loads so it does not evict
// the L2-resident xw/aggw working set. xw stores stay RT (reused by scatter).
// ---------------------------------------------------------------------------
__global__ void node_proj_kernel(const float* __restrict__ x,
                                 const float* __restrict__ w,
                                 float* __restrict__ xw,
                                 int n_nodes) {
    __shared__ float sW[16 * WSTRIDE];
    const int tid = threadIdx.x;
    for (int i = tid; i < 16 * IN_CH; i += blockDim.x)
        sW[(i >> 7) * WSTRIDE + (i & 127)] = w[i];
    __syncthreads();

    const int wave = tid >> 5;
    const int lane = tid & 31;
    const int hi   = lane >> 4;   // 0: K pair {0,1}, 1: K pair {2,3}
    const int lj   = lane & 15;
    const int base = (blockIdx.x * 8 + wave) * 16;

    int arow = base + lj;
    if (arow > n_nodes - 1) arow = n_nodes - 1;   // clamp loads; stores guarded below
    const float* aptr = x + (size_t)arow * IN_CH + hi * 2;
    const float* bptr = &sW[lj * WSTRIDE + hi * 2];

    v8f c = {};
#pragma unroll
    for (int k0 = 0; k0 < IN_CH; k0 += 4) {
        v2f a = __builtin_nontemporal_load((const v2f*)(aptr + k0));
        v2f b = *(const v2f*)(bptr + k0);
        c = __builtin_amdgcn_wmma_f32_16x16x4_f32(false, a, false, b, (short)0, c,
                                                  false, false);
    }
#pragma unroll
    for (int r = 0; r < 8; ++r) {
        int node = base + r + hi * 8;          // C layout: M = r (+8 for hi lanes)
        if (node < n_nodes) xw[(size_t)node * OUT_CH + lj] = c[r];
    }
}

// ---------------------------------------------------------------------------
// Kernel 2: zero the aggregation buffer (graph-capture safe, deterministic)
// ---------------------------------------------------------------------------
__global__ void zero_kernel(float4* __restrict__ p, int n4) {
    int i = blockIdx.x * blockDim.x + threadIdx.x;
    if (i < n4) p[i] = make_float4(0.f, 0.f, 0.f, 0.f);
}

// ---------------------------------------------------------------------------
// Kernel 3: aggw[dst[e]] += xw[src[e]]  (16 floats/edge, 4 threads/edge).
// xw (32MB) and aggw (6.4MB) are both L2-resident -> gather hits L2 and the
// 51M f32 atomics resolve in L2 without HBM round-trips. Keep RT hints.
// ---------------------------------------------------------------------------
__global__ void edge_scatter_kernel(const int* __restrict__ edge_index,
                                    const float* __restrict__ xw,
                                    float* __restrict__ aggw,
                                    int E) {
    int gid  = blockIdx.x * blockDim.x + threadIdx.x;
    int e    = gid >> 2;
    int part = gid & 3;
    if (e >= E) return;
    int s = edge_index[e];        // row 0: src
    int d = edge_index[E + e];    // row 1: dst
    const float4 v = *(const float4*)(xw + (size_t)s * OUT_CH + part * 4);
    float* o = aggw + (size_t)d * OUT_CH + part * 4;
    atomicAdd(o + 0, v.x);
    atomicAdd(o + 1, v.y);
    atomicAdd(o + 2, v.z);
    atomicAdd(o + 3, v.w);
}

// ---------------------------------------------------------------------------
// Kernel 4: fused epilogue per 16-node tile (one wave each):
//   h  = aggw + b_l + x[:B] @ w_r.T ; leaky
//   h1 = leaky(h @ w1.T + b1)
//   out = h1 @ w2.T + b2
// 16x16 intermediates round-trip through a per-wave LDS tile to rebuild the
// A-fragment layout (same-wave DS ops are in-order; no barrier needed).
// x reads are single-pass (NT); out stores are never re-read (NT).
// ---------------------------------------------------------------------------
__global__ void mlp_kernel(const float* __restrict__ x,
                           const float* __restrict__ aggw,
                           const float* __restrict__ w_r,
                           const float* __restrict__ b_l,
                           const float* __restrict__ w1,
                           const float* __restrict__ b1,
                           const float* __restrict__ w2,
                           const float* __restrict__ b2,
                           float* __restrict__ out,
                           int B) {
    __shared__ float sWr[16 * WSTRIDE];
    __shared__ float sW1[16 * HSTRIDE];
    __shared__ float sW2[16 * HSTRIDE];
    __shared__ float sBl[16], sB1[16], sB2[16];
    __shared__ float sH[8][16 * HSTRIDE];     // per-wave 16x16 staging tiles

    const int tid = threadIdx.x;
    for (int i = tid; i < 16 * IN_CH; i += blockDim.x)
        sWr[(i >> 7) * WSTRIDE + (i & 127)] = w_r[i];
    for (int i = tid; i < 256; i += blockDim.x) {
        sW1[(i >> 4) * HSTRIDE + (i & 15)] = w1[i];
        sW2[(i >> 4) * HSTRIDE + (i & 15)] = w2[i];
    }
    if (tid < 16) { sBl[tid] = b_l[tid]; sB1[tid] = b1[tid]; sB2[tid] = b2[tid]; }
    __syncthreads();

    const int wave = tid >> 5;
    const int lane = tid & 31;
    const int hi   = lane >> 4;
    const int lj   = lane & 15;
    const int base = (blockIdx.x * 8 + wave) * 16;

    int arow = base + lj;
    if (arow > B - 1) arow = B - 1;
    const float* aptr = x + (size_t)arow * IN_CH + hi * 2;
    const float* bptr = &sWr[lj * WSTRIDE + hi * 2];

    // x_target @ w_r.T
    v8f c = {};
#pragma unroll
    for (int k0 = 0; k0 < IN_CH; k0 += 4) {
        v2f a = __builtin_nontemporal_load((const v2f*)(aptr + k0));
        v2f b = *(const v2f*)(bptr + k0);
        c = __builtin_amdgcn_wmma_f32_16x16x4_f32(false, a, false, b, (short)0, c,
                                                  false, false);
    }

    // h = aggw + b_l + x@w_r.T ; leaky ; stage M-major in LDS
    const float bl = sBl[lj];
    float* hT = sH[wave];
#pragma unroll
    for (int r = 0; r < 8; ++r) {
        int node = base + r + hi * 8;
        int cn   = node > B - 1 ? B - 1 : node;
        float h  = c[r] + aggw[(size_t)cn * OUT_CH + lj] + bl;
        hT[(r + hi * 8) * HSTRIDE + lj] = leaky(h);
    }

    // h1 = leaky(h @ w1.T + b1)
    v8f c2 = {};
#pragma unroll
    for (int k0 = 0; k0 < 16; k0 += 4) {
        v2f a = *(const v2f*)(&hT[lj * HSTRIDE + k0 + hi * 2]);
        v2f b = *(const v2f*)(&sW1[lj * HSTRIDE + k0 + hi * 2]);
        c2 = __builtin_amdgcn_wmma_f32_16x16x4_f32(false, a, false, b, (short)0, c2,
                                                   false, false);
    }
    const float bb1 = sB1[lj];
#pragma unroll
    for (int r = 0; r < 8; ++r)
        hT[(r + hi * 8) * HSTRIDE + lj] = leaky(c2[r] + bb1);

    // out = h1 @ w2.T + b2
    v8f c3 = {};
#pragma unroll
    for (int k0 = 0; k0 < 16; k0 += 4) {
        v2f a = *(const v2f*)(&hT[lj * HSTRIDE + k0 + hi * 2]);
        v2f b = *(const v2f*)(&sW2[lj * HSTRIDE + k0 + hi * 2]);
        c3 = __builtin_amdgcn_wmma_f32_16x16x4_f32(false, a, false, b, (short)0, c3,
                                                   false, false);
    }
    const float bb2 = sB2[lj];
#pragma unroll
    for (int r = 0; r < 8; ++r) {
        int node = base + r + hi * 8;
        if (node < B)
            __builtin_nontemporal_store(c3[r] + bb2, &out[(size_t)node * OUT_CH + lj]);
    }
}

// ---------------------------------------------------------------------------
extern "C" void kernel_launch(void* const* d_in, const int* in_sizes, int n_in,
                              void* d_out, int out_size, void* d_ws, size_t ws_size,
                              hipStream_t stream) {
    const float* x   = (const float*)d_in[0];
    const int*   ei  = (const int*)d_in[1];   // edge_index (2, E)
    const float* w_l = (const float*)d_in[3];
    const float* b_l = (const float*)d_in[4];
    const float* w_r = (const float*)d_in[5];
    const float* w1  = (const float*)d_in[6];
    const float* b1  = (const float*)d_in[7];
    const float* w2  = (const float*)d_in[8];
    const float* b2  = (const float*)d_in[9];
    float* out = (float*)d_out;

    const int n_nodes = in_sizes[0] / IN_CH;
    const int E       = in_sizes[1] / 2;
    const int B       = out_size / OUT_CH;    // derive batch_size host-side

    // workspace: xw[n_nodes*16] | aggw[B*16]  (32MB + 6.4MB)
    float* xw   = (float*)d_ws;
    float* aggw = xw + (size_t)n_nodes * OUT_CH;

    int n4 = (B * OUT_CH) / 4;
    zero_kernel<<<(n4 + 255) / 256, 256, 0, stream>>>((float4*)aggw, n4);

    node_proj_kernel<<<(n_nodes + 127) / 128, 256, 0, stream>>>(x, w_l, xw, n_nodes);

    long long tot = (long long)E * 4;
    edge_scatter_kernel<<<(int)((tot + 255) / 256), 256, 0, stream>>>(ei, xw, aggw, E);

    mlp_kernel<<<(B + 127) / 128, 256, 0, stream>>>(x, aggw, w_r, b_l, w1, b1, w2, b2,
                                                    out, B);
}